// TypeGAT_60756607369366
// MI455X (gfx1250) — compile-verified
//
#include <hip/hip_runtime.h>

typedef float v2f __attribute__((ext_vector_type(2)));
typedef float v8f __attribute__((ext_vector_type(8)));

#define NEGF (-1e30f)
#define CAP 1024     // per-list candidate cap (expected ~13 matches/entity)
#define MAXR 512

// ---------------------------------------------------------------------------
// threefry-2x32 (JAX default PRNG), key = (0, 42) from jax.random.key(42).
// Replicates random_bits layout: counts iota(B) split into halves
// x0 = [0..B/2), x1 = [B/2..B); output = concat(y0, y1).
// ---------------------------------------------------------------------------
__device__ __forceinline__ unsigned rotl32(unsigned x, int n) {
    return (x << n) | (x >> (32 - n));
}

__device__ unsigned threefry_bits(unsigned idx, unsigned half_n) {
    unsigned x0, x1;
    bool first;
    if (idx < half_n) { x0 = idx;          x1 = idx + half_n; first = true;  }
    else              { x0 = idx - half_n; x1 = idx;          first = false; }
    const unsigned k0 = 0u, k1 = 42u;
    const unsigned k2 = k0 ^ k1 ^ 0x1BD11BDAu;
    const unsigned ks[3] = {k0, k1, k2};
    const int R0[4] = {13, 15, 26, 6};
    const int R1[4] = {17, 29, 16, 24};
    x0 += ks[0]; x1 += ks[1];
#pragma unroll
    for (int g = 0; g < 5; ++g) {
        const int* r = (g & 1) ? R1 : R0;
#pragma unroll
        for (int q = 0; q < 4; ++q) { x0 += x1; x1 = rotl32(x1, r[q]); x1 ^= x0; }
        x0 += ks[(g + 1) % 3];
        x1 += ks[(g + 2) % 3] + (unsigned)(g + 1);
    }
    return first ? x0 : x1;
}

// ---------------------------------------------------------------------------
// Phase 1: one workgroup per entity. Stream edges (b128 loads) from L2,
// collect incident-edge candidates in LDS, run iterative block-argmax top-k,
// emit a relation-count histogram row + divisor + fallback metadata.
// ---------------------------------------------------------------------------
__global__ __launch_bounds__(256) void hics_select_kernel(
    const int* __restrict__ entity_index,
    const int* __restrict__ edge_src,
    const int* __restrict__ edge_dst,
    const int* __restrict__ edge_types,
    const float* __restrict__ edge_times,
    const int* __restrict__ d_k,
    const int* __restrict__ d_num_nodes,
    float* __restrict__ counts,   // [B, R] relation histogram (f32)
    float* __restrict__ inv_cnt,  // [B]
    int* __restrict__ no_nb,      // [B]
    int* __restrict__ vrel,       // [B]
    int B, int E, int R)
{
    __shared__ float s_time[2][CAP];
    __shared__ int   s_type[2][CAP];
    __shared__ int   s_cnt[2];
    __shared__ float red_val[256];
    __shared__ int   red_idx[256];
    __shared__ float s_row[MAXR];

    const int tid = threadIdx.x;
    const int b   = blockIdx.x;
    const int k   = *d_k;            // scalar input lives on device
    const int nn  = *d_num_nodes;
    const int kh  = k >> 1;

    if (tid < 2) s_cnt[tid] = 0;
    for (int r = tid; r < R; r += 256) s_row[r] = 0.0f;
    __syncthreads();

    const int  ent   = entity_index[b];
    const bool valid = (ent >= 0) && (ent < nn);

    // -------- edge scan: 4 edges per thread-iteration (global_load_b128) ----
    const int  E4   = E >> 2;
    const int4* s4  = (const int4*)edge_src;
    const int4* dd4 = (const int4*)edge_dst;
    for (int i = tid; i < E4; i += 256) {
        const int4 s = s4[i];
        const int4 d = dd4[i];
        const int  e = i << 2;
        if (d.x == ent) { int p = atomicAdd(&s_cnt[0], 1); if (p < CAP) { s_time[0][p] = edge_times[e+0]; s_type[0][p] = edge_types[e+0]; } }
        if (d.y == ent) { int p = atomicAdd(&s_cnt[0], 1); if (p < CAP) { s_time[0][p] = edge_times[e+1]; s_type[0][p] = edge_types[e+1]; } }
        if (d.z == ent) { int p = atomicAdd(&s_cnt[0], 1); if (p < CAP) { s_time[0][p] = edge_times[e+2]; s_type[0][p] = edge_types[e+2]; } }
        if (d.w == ent) { int p = atomicAdd(&s_cnt[0], 1); if (p < CAP) { s_time[0][p] = edge_times[e+3]; s_type[0][p] = edge_types[e+3]; } }
        if (s.x == ent) { int p = atomicAdd(&s_cnt[1], 1); if (p < CAP) { s_time[1][p] = edge_times[e+0]; s_type[1][p] = edge_types[e+0]; } }
        if (s.y == ent) { int p = atomicAdd(&s_cnt[1], 1); if (p < CAP) { s_time[1][p] = edge_times[e+1]; s_type[1][p] = edge_types[e+1]; } }
        if (s.z == ent) { int p = atomicAdd(&s_cnt[1], 1); if (p < CAP) { s_time[1][p] = edge_times[e+2]; s_type[1][p] = edge_types[e+2]; } }
        if (s.w == ent) { int p = atomicAdd(&s_cnt[1], 1); if (p < CAP) { s_time[1][p] = edge_times[e+3]; s_type[1][p] = edge_types[e+3]; } }
    }
    for (int e = (E4 << 2) + tid; e < E; e += 256) {  // tail (E%4 != 0)
        if (edge_dst[e] == ent) { int p = atomicAdd(&s_cnt[0], 1); if (p < CAP) { s_time[0][p] = edge_times[e]; s_type[0][p] = edge_types[e]; } }
        if (edge_src[e] == ent) { int p = atomicAdd(&s_cnt[1], 1); if (p < CAP) { s_time[1][p] = edge_times[e]; s_type[1][p] = edge_types[e]; } }
    }
    __syncthreads();

    const int inc_cnt  = s_cnt[0];
    const int out_cnt  = s_cnt[1];
    const int inc_take = min(kh, inc_cnt);
    const int rem      = (inc_take > 0) ? (k - inc_take) : k;
    const int out_take = min(rem, out_cnt);

    // -------- iterative top-k by block-wide argmax over LDS candidates ------
    for (int l = 0; l < 2; ++l) {
        const int take = (l == 0) ? inc_take : out_take;
        const int len  = min((l == 0) ? inc_cnt : out_cnt, CAP);
        for (int j = 0; j < take; ++j) {
            float best = NEGF; int bi = -1;
            for (int i = tid; i < len; i += 256) {
                const float t = s_time[l][i];
                if (t > best) { best = t; bi = i; }
            }
            red_val[tid] = best; red_idx[tid] = bi;
            __syncthreads();
            for (int s = 128; s > 0; s >>= 1) {
                if (tid < s && red_val[tid + s] > red_val[tid]) {
                    red_val[tid] = red_val[tid + s];
                    red_idx[tid] = red_idx[tid + s];
                }
                __syncthreads();
            }
            if (tid == 0) {
                const int w = red_idx[0];
                if (w >= 0) { s_row[s_type[l][w]] += 1.0f; s_time[l][w] = NEGF; }
            }
            __syncthreads();
        }
    }

    if (tid == 0) {
        const int c = inc_take + out_take;
        inv_cnt[b] = 1.0f / fmaxf((float)c, 1.0f);
        no_nb[b]   = (((inc_cnt == 0) && (out_cnt == 0)) || !valid) ? 1 : 0;
        const unsigned bits = threefry_bits((unsigned)b, (unsigned)(B >> 1));
        vrel[b] = (int)(bits % (unsigned)R);
    }
    __syncthreads();
    for (int r = tid; r < R; r += 256) counts[(size_t)b * R + r] = s_row[r];
}

// ---------------------------------------------------------------------------
// Phase 2: summed[B,D] = counts[B,R] x query_weight[R,D] on the WMMA pipe
// (V_WMMA_F32_16X16X4_F32), fused with 1/cnt scaling + virtual fallback.
// One wave per 16x16 output tile; a block's 8 waves share one M-tile row
// of counts (hot in L0/WGP$).
// ---------------------------------------------------------------------------
__global__ __launch_bounds__(256) void hics_wmma_agg_kernel(
    const float* __restrict__ counts,   // [M, K]
    const float* __restrict__ qw,       // [K, N]
    const float* __restrict__ inv_cnt,  // [M]
    const int* __restrict__ no_nb,      // [M]
    const int* __restrict__ vrel,       // [M]
    float* __restrict__ out,            // [M, N]
    int M, int K, int N)
{
    const int tid    = threadIdx.x;
    const int lane   = tid & 31;
    const int wave   = tid >> 5;
    const int tilesN = N >> 4;
    const int tile   = blockIdx.x * 8 + wave;
    const int tileM  = tile / tilesN;
    const int tileN  = tile % tilesN;
    if (tileM * 16 >= M) return;   // uniform per wave

    const int half = lane >> 4;    // wave32 half-select (K split)
    const int l    = lane & 15;
    const int m0   = tileM * 16, n0 = tileN * 16;

    // A (16x4 f32): lanes 0-15 -> K=0,1 ; lanes 16-31 -> K=2,3
    // B (4x16 f32): v0/v1 lanes 0-15 -> K=0/1 ; lanes 16-31 -> K=2/3
    const float* arow = counts + (size_t)(m0 + l) * K + half * 2;
    const float* bcol = qw + (size_t)(half * 2) * N + n0 + l;

    v8f c = {};
    for (int kk = 0; kk < K; kk += 4) {
        v2f a, bm;
        a.x  = arow[kk];
        a.y  = arow[kk + 1];
        bm.x = bcol[(size_t)kk * N];
        bm.y = bcol[(size_t)kk * N + N];
        c = __builtin_amdgcn_wmma_f32_16x16x4_f32(
                /*neg_a=*/false, a, /*neg_b=*/false, bm,
                /*c_mod=*/(short)0, c, /*reuse_a=*/false, /*reuse_b=*/false);
    }

    const float decay = 0.90483741803595957f;  // exp(-0.1)
#pragma unroll
    for (int i = 0; i < 8; ++i) {
        // C/D layout: lanes 0-15 -> M=i, lanes 16-31 -> M=i+8, col = lane&15
        const int row = m0 + half * 8 + i;
        const int col = n0 + l;
        float v = c[i] * inv_cnt[row];
        if (no_nb[row]) v = qw[(size_t)vrel[row] * N + col] * decay;
        out[(size_t)row * N + col] = v;
    }
}

// ---------------------------------------------------------------------------
extern "C" void kernel_launch(void* const* d_in, const int* in_sizes, int n_in,
                              void* d_out, int out_size, void* d_ws, size_t ws_size,
                              hipStream_t stream) {
    const int*   entity_index = (const int*)d_in[0];
    const int*   edge_src     = (const int*)d_in[1];
    const int*   edge_dst     = (const int*)d_in[2];
    const int*   edge_types   = (const int*)d_in[3];
    const float* edge_times   = (const float*)d_in[4];
    const float* qw           = (const float*)d_in[5];
    const int*   d_k          = (const int*)d_in[6];
    const int*   d_num_nodes  = (const int*)d_in[7];

    const int B = in_sizes[0];
    const int E = in_sizes[1];
    const int D = out_size / B;         // 128
    const int R = in_sizes[5] / D;      // 256

    // scratch layout
    float* counts  = (float*)d_ws;                       // B*R floats
    float* inv_cnt = counts + (size_t)B * R;             // B floats
    int*   no_nb   = (int*)(inv_cnt + B);                // B ints
    int*   vrel    = no_nb + B;                          // B ints

    hics_select_kernel<<<B, 256, 0, stream>>>(
        entity_index, edge_src, edge_dst, edge_types, edge_times,
        d_k, d_num_nodes, counts, inv_cnt, no_nb, vrel, B, E, R);

    const int tiles  = (B / 16) * (D / 16);
    const int blocks = (tiles + 7) / 8;
    hics_wmma_agg_kernel<<<blocks, 256, 0, stream>>>(
        counts, qw, inv_cnt, no_nb, vrel, (float*)d_out, B, R, D);
}